// GINMolHeadEncoder_9251359555631
// MI455X (gfx1250) — compile-verified
//
#include <hip/hip_runtime.h>

#define DIM 128
#define DIM2 256
#define NLAYER 5
#define NF_A 9
#define VA 119
#define NF_B 3
#define VB 6

typedef float v2f __attribute__((ext_vector_type(2)));
typedef float v8f __attribute__((ext_vector_type(8)));

__device__ __forceinline__ void atomAddF(float* p, float v) {
    unsafeAtomicAdd(p, v);   // native global_atomic_add_f32 on CDNA5
}

// ---------------- Atom encoder: h[n,d] = sum_f atom_emb[f][x[n,f]][d] ----------------
__global__ __launch_bounds__(128) void k_atom_encoder(const int* __restrict__ x,
                                                      const float* __restrict__ aemb,
                                                      float* __restrict__ h, int n) {
    int node = blockIdx.x;
    if (node >= n) return;
    int d = threadIdx.x;
    float acc = 0.f;
#pragma unroll
    for (int f = 0; f < NF_A; ++f) {
        int xi = x[node * NF_A + f];
        acc += aemb[((size_t)(f * VA + xi)) * DIM + d];
    }
    h[(size_t)node * DIM + d] = acc;
}

// ---------------- aggr = (1+eps[l]) * h ----------------
__global__ __launch_bounds__(256) void k_init_aggr(const float* __restrict__ h,
                                                   const float* __restrict__ eps, int l,
                                                   float* __restrict__ aggr, int count) {
    int i = blockIdx.x * 256 + threadIdx.x;
    if (i < count) aggr[i] = (1.0f + eps[l]) * h[i];
}

// ---------------- message + scatter: aggr[dst] += relu(h[src] + bond_emb) ----------------
__global__ __launch_bounds__(256) void k_message_scatter(const int* __restrict__ ei,
                                                         const int* __restrict__ ea,
                                                         const float* __restrict__ bemb,
                                                         const float* __restrict__ h,
                                                         float* __restrict__ aggr, int E) {
    int e = blockIdx.x * 2 + (threadIdx.x >> 7);
    int d = threadIdx.x & (DIM - 1);
    if (e >= E) return;
    int src = ei[e];
    int dst = ei[E + e];
    float v = h[(size_t)src * DIM + d];
#pragma unroll
    for (int f = 0; f < NF_B; ++f) {
        int bv = ea[e * NF_B + f];
        v += bemb[(f * VB + bv) * DIM + d];
    }
    v = fmaxf(v, 0.f);
    atomAddF(&aggr[(size_t)dst * DIM + d], v);
}

// ---------------- zero BN stat buffers ----------------
__global__ void k_zero_stats(float* __restrict__ ssum, float* __restrict__ ssq) {
    ssum[threadIdx.x] = 0.f;
    ssq[threadIdx.x] = 0.f;
}

// ---------------- weight transpose: Wt[c][k] = W[k][c] (weights are tiny) ----------------
__global__ __launch_bounds__(256) void k_transpose(const float* __restrict__ W,
                                                   float* __restrict__ Wt, int K, int Nc) {
    int i = blockIdx.x * 256 + threadIdx.x;
    if (i >= K * Nc) return;
    int k = i / Nc, c = i % Nc;
    Wt[(size_t)c * K + k] = W[i];
}

// ---------------- GEMM1: z1[N,256] = aggr[N,128] @ W1 + b1 (fp32 WMMA) ----------------
// 8 waves x 16 cols; each wave computes 2 M-subtiles (32 rows) reusing B fragments.
// A tile staged via CDNA5 async global->LDS copy. B from transposed weights (b64 loads).
__global__ __launch_bounds__(256) void k_gemm1(const float* __restrict__ A,
                                               const float* __restrict__ Wt, // [256][128]
                                               const float* __restrict__ bias,
                                               float* __restrict__ Z, int nrows) {
    __shared__ float As[32 * 132];   // 32 rows x 128 cols, stride 132 (bank pad)
    int t = threadIdx.x;
    int lane = t & 31;
    int wave = t >> 5;               // 0..7 -> 16-column slice
    int m0 = blockIdx.x * 32;
    int n0 = blockIdx.y * 128;

    // async stage A tile (32x128 fp32) into LDS: 4 x b128 per thread
    {
        int c4 = (t & 31) * 4;       // 0..124
        int rb = t >> 5;             // 0..7
#pragma unroll
        for (int i = 0; i < 4; ++i) {
            int r = rb + i * 8;
            int gr = m0 + r;
            if (gr >= nrows) gr = nrows - 1;   // clamp: tail rows never stored
            const float* gp = A + (size_t)gr * DIM + c4;
            unsigned int lp = (unsigned int)(uintptr_t)(&As[r * 132 + c4]);
            asm volatile("global_load_async_to_lds_b128 %0, %1, off"
                         :: "v"(lp), "v"((unsigned long long)(uintptr_t)gp)
                         : "memory");
        }
        asm volatile("s_wait_asynccnt 0x0" ::: "memory");
    }
    __syncthreads();

    v8f acc0 = {}, acc1 = {};
    int rlo   = lane & 15;
    int khalf = (lane >> 4) * 2;                 // 0 or 2
    int col   = n0 + wave * 16 + rlo;            // global output column
    const float* Bcol = Wt + (size_t)col * DIM;  // contiguous K for this column
    const float* Ar0 = &As[rlo * 132];
    const float* Ar1 = &As[(rlo + 16) * 132];

    for (int k0 = 0; k0 < DIM; k0 += 4) {
        v2f b = *(const v2f*)(Bcol + k0 + khalf);     // one global_load_b64
        v2f a0, a1;
        a0.x = Ar0[k0 + khalf];
        a0.y = Ar0[k0 + khalf + 1];
        a1.x = Ar1[k0 + khalf];
        a1.y = Ar1[k0 + khalf + 1];
        acc0 = __builtin_amdgcn_wmma_f32_16x16x4_f32(false, a0, false, b, (short)0, acc0,
                                                     false, false);
        acc1 = __builtin_amdgcn_wmma_f32_16x16x4_f32(false, a1, false, b, (short)0, acc1,
                                                     false, false);
    }

    float bv = bias[col];
    int grb = m0 + ((lane >> 4) * 8);
#pragma unroll
    for (int r = 0; r < 8; ++r) {
        int gr = grb + r;
        if (gr < nrows)      Z[(size_t)gr * DIM2 + col]        = acc0[r] + bv;
        if (gr + 16 < nrows) Z[(size_t)(gr + 16) * DIM2 + col] = acc1[r] + bv;
    }
}

// ---------------- per-column sum / sumsq over rows ----------------
__global__ __launch_bounds__(256) void k_colstats(const float* __restrict__ z, int nrows,
                                                  int ncols, float* __restrict__ ssum,
                                                  float* __restrict__ ssq) {
    int t = threadIdx.x;
    int c = t % ncols;
    int ro = t / ncols;
    int stride = 256 / ncols;
    int r0 = blockIdx.x * 64;
    float s = 0.f, sq = 0.f;
    for (int r = ro; r < 64; r += stride) {
        int gr = r0 + r;
        if (gr < nrows) {
            float v = z[(size_t)gr * ncols + c];
            s += v;
            sq += v * v;
        }
    }
    atomAddF(&ssum[c], s);
    atomAddF(&ssq[c], sq);
}

// ---------------- fold mean/var + gamma/beta into scale/shift ----------------
__global__ void k_bn_params(const float* __restrict__ ssum, const float* __restrict__ ssq,
                            const float* __restrict__ g, const float* __restrict__ b,
                            float* __restrict__ scale, float* __restrict__ shift,
                            int nrows, int ncols) {
    int c = threadIdx.x;
    if (c >= ncols) return;
    float inv_n = 1.0f / (float)nrows;
    float mean = ssum[c] * inv_n;
    float var = ssq[c] * inv_n - mean * mean;
    float rstd = rsqrtf(var + 1e-5f);
    float sc = rstd * g[c];
    scale[c] = sc;
    shift[c] = b[c] - mean * sc;
}

// ---------------- GEMM2: z2[N,128] = relu(bn1(z1)) @ W2 + b2 (fp32 WMMA) ----------------
// BN1+ReLU fused into synchronous A-tile staging; same 8x16-col / 2-Mtile wave layout.
__global__ __launch_bounds__(256) void k_gemm2(const float* __restrict__ Z1,
                                               const float* __restrict__ scale,
                                               const float* __restrict__ shift,
                                               const float* __restrict__ Wt, // [128][256]
                                               const float* __restrict__ bias,
                                               float* __restrict__ Z2, int nrows) {
    __shared__ float As[32 * 260];   // 32 rows x 256 cols, stride 260 (bank pad)
    int t = threadIdx.x;
    int lane = t & 31;
    int wave = t >> 5;
    int m0 = blockIdx.x * 32;

    // stage BN1+ReLU(A) tile into LDS
    {
        int c4 = (t & 63) * 4;       // 0..252
        int rb = t >> 6;             // 0..3
        float4 s4 = *(const float4*)(scale + c4);
        float4 h4 = *(const float4*)(shift + c4);
#pragma unroll
        for (int i = 0; i < 8; ++i) {
            int r = rb + i * 4;
            int gr = m0 + r;
            float4 v = make_float4(0.f, 0.f, 0.f, 0.f);
            if (gr < nrows) {
                float4 z = *(const float4*)(Z1 + (size_t)gr * DIM2 + c4);
                v.x = fmaxf(z.x * s4.x + h4.x, 0.f);
                v.y = fmaxf(z.y * s4.y + h4.y, 0.f);
                v.z = fmaxf(z.z * s4.z + h4.z, 0.f);
                v.w = fmaxf(z.w * s4.w + h4.w, 0.f);
            }
            *(float4*)(&As[r * 260 + c4]) = v;
        }
    }
    __syncthreads();

    v8f acc0 = {}, acc1 = {};
    int rlo   = lane & 15;
    int khalf = (lane >> 4) * 2;
    int col   = wave * 16 + rlo;                  // 8 waves cover all 128 columns
    const float* Bcol = Wt + (size_t)col * DIM2;
    const float* Ar0 = &As[rlo * 260];
    const float* Ar1 = &As[(rlo + 16) * 260];

    for (int k0 = 0; k0 < DIM2; k0 += 4) {
        v2f b = *(const v2f*)(Bcol + k0 + khalf);
        v2f a0, a1;
        a0.x = Ar0[k0 + khalf];
        a0.y = Ar0[k0 + khalf + 1];
        a1.x = Ar1[k0 + khalf];
        a1.y = Ar1[k0 + khalf + 1];
        acc0 = __builtin_amdgcn_wmma_f32_16x16x4_f32(false, a0, false, b, (short)0, acc0,
                                                     false, false);
        acc1 = __builtin_amdgcn_wmma_f32_16x16x4_f32(false, a1, false, b, (short)0, acc1,
                                                     false, false);
    }

    float bv = bias[col];
    int grb = m0 + ((lane >> 4) * 8);
#pragma unroll
    for (int r = 0; r < 8; ++r) {
        int gr = grb + r;
        if (gr < nrows)      Z2[(size_t)gr * DIM + col]        = acc0[r] + bv;
        if (gr + 16 < nrows) Z2[(size_t)(gr + 16) * DIM + col] = acc1[r] + bv;
    }
}

// ---------------- h_new = bn2(z2) (+relu) + h ----------------
__global__ __launch_bounds__(256) void k_finalize(const float* __restrict__ z2,
                                                  const float* __restrict__ scale,
                                                  const float* __restrict__ shift,
                                                  const float* __restrict__ h,
                                                  float* __restrict__ dst, int count,
                                                  int do_relu) {
    int i = blockIdx.x * 256 + threadIdx.x;
    if (i >= count) return;
    int d = i & (DIM - 1);
    float v = z2[i] * scale[d] + shift[d];
    if (do_relu) v = fmaxf(v, 0.f);
    dst[i] = v + h[i];
}

extern "C" void kernel_launch(void* const* d_in, const int* in_sizes, int n_in,
                              void* d_out, int out_size, void* d_ws, size_t ws_size,
                              hipStream_t stream) {
    const int* x          = (const int*)d_in[0];
    const int* edge_index = (const int*)d_in[1];
    const int* edge_attr  = (const int*)d_in[2];
    /* d_in[3] batch — unused by reference output */
    const float* atom_emb = (const float*)d_in[4];
    const float* bond_emb = (const float*)d_in[5];
    const float* eps      = (const float*)d_in[6];
    const float* W1       = (const float*)d_in[7];
    const float* b1       = (const float*)d_in[8];
    const float* bn1_g    = (const float*)d_in[9];
    const float* bn1_b    = (const float*)d_in[10];
    const float* W2       = (const float*)d_in[11];
    const float* b2       = (const float*)d_in[12];
    const float* bn_g     = (const float*)d_in[13];
    const float* bn_b     = (const float*)d_in[14];

    int n = in_sizes[0] / NF_A;
    int e = in_sizes[1] / 2;

    float* h    = (float*)d_ws;
    float* aggr = h + (size_t)n * DIM;        // also reused as z2
    float* z1   = aggr + (size_t)n * DIM;
    float* ssum = z1 + (size_t)n * DIM2;
    float* ssq  = ssum + DIM2;
    float* scl  = ssq + DIM2;
    float* shf  = scl + DIM2;
    float* Wt1  = shf + DIM2;                 // [256][128]
    float* Wt2  = Wt1 + DIM * DIM2;           // [128][256]

    int cnt = n * DIM;
    int ew_blocks = (cnt + 255) / 256;
    int wt_blocks = (DIM * DIM2 + 255) / 256;

    k_atom_encoder<<<n, 128, 0, stream>>>(x, atom_emb, h, n);

    for (int l = 0; l < NLAYER; ++l) {
        k_transpose<<<wt_blocks, 256, 0, stream>>>(W1 + (size_t)l * DIM * DIM2, Wt1,
                                                   DIM, DIM2);
        k_transpose<<<wt_blocks, 256, 0, stream>>>(W2 + (size_t)l * DIM2 * DIM, Wt2,
                                                   DIM2, DIM);

        k_init_aggr<<<ew_blocks, 256, 0, stream>>>(h, eps, l, aggr, cnt);
        k_message_scatter<<<(e + 1) / 2, 256, 0, stream>>>(
            edge_index, edge_attr, bond_emb + (size_t)l * NF_B * VB * DIM, h, aggr, e);

        // GEMM1 + BN1 stats
        k_zero_stats<<<1, 512, 0, stream>>>(ssum, ssq);
        dim3 g1((n + 31) / 32, DIM2 / 128);
        k_gemm1<<<g1, 256, 0, stream>>>(aggr, Wt1, b1 + (size_t)l * DIM2, z1, n);
        k_colstats<<<(n + 63) / 64, 256, 0, stream>>>(z1, n, DIM2, ssum, ssq);
        k_bn_params<<<1, 256, 0, stream>>>(ssum, ssq, bn1_g + (size_t)l * DIM2,
                                           bn1_b + (size_t)l * DIM2, scl, shf, n, DIM2);

        // GEMM2 (BN1+ReLU fused into A staging) + BN2 stats
        k_gemm2<<<(n + 31) / 32, 256, 0, stream>>>(z1, scl, shf, Wt2,
                                                   b2 + (size_t)l * DIM, aggr, n);
        k_zero_stats<<<1, 512, 0, stream>>>(ssum, ssq);
        k_colstats<<<(n + 63) / 64, 256, 0, stream>>>(aggr, n, DIM, ssum, ssq);
        k_bn_params<<<1, 128, 0, stream>>>(ssum, ssq, bn_g + (size_t)l * DIM,
                                           bn_b + (size_t)l * DIM, scl, shf, n, DIM);

        float* dst = (l == NLAYER - 1) ? (float*)d_out : h;
        k_finalize<<<ew_blocks, 256, 0, stream>>>(aggr, scl, shf, h, dst, cnt,
                                                  (l < NLAYER - 1) ? 1 : 0);
    }
}